// BMCLoss_26242250179443
// MI455X (gfx1250) — compile-verified
//
#include <hip/hip_runtime.h>
#include <hip/hip_bf16.h>

typedef float v2f __attribute__((ext_vector_type(2)));
typedef float v8f __attribute__((ext_vector_type(8)));
typedef unsigned int v4u __attribute__((ext_vector_type(4)));
typedef int v4i __attribute__((ext_vector_type(4)));
typedef int v8i __attribute__((ext_vector_type(8)));

#define B_TOT   8192
#define D_TOT   256
#define MBLK    128
#define NBLK    64
#define SPLITS  16
#define COLS_PER_WG (B_TOT / SPLITS)   // 512
#define NITER   (COLS_PER_WG / NBLK)   // 8
#define PADROW  (D_TOT + 4)            // 260 floats/row in LDS (bank-conflict-free b64 reads)
#define LOG2E   1.4426950408889634f
#define LN2     0.6931471805599453f

// LDS byte offsets (dynamic LDS base == 0: kernel has no static __shared__)
#define LDSA_OFF  0u
#define LDSB0_OFF ((unsigned)(MBLK * PADROW * 4))                     // 133120
#define LDSB1_OFF ((unsigned)((MBLK + NBLK) * PADROW * 4))            // 199680
#define LDS_TOTAL ((size_t)(MBLK + 2 * NBLK) * PADROW * 4)            // 266240 B < 320 KB

__device__ __forceinline__ float fexp2(float x) { return __builtin_amdgcn_exp2f(x); }

// ---------------------------------------------------------------------------
// TDM: DMA a [tileRows x 256] f32 row-major tile into LDS, inserting 16 B of
// padding after every 1024 B row (pad_interval=7 -> 256 DWORDs, pad_amount=3
// -> 4 DWORDs) so the LDS image has 260-float rows. Tracked on TENSORcnt.
// ---------------------------------------------------------------------------
__device__ __forceinline__ void tdm_load_2d_to_lds(unsigned ldsOff, const void* gsrc,
                                                   unsigned tileRows) {
    unsigned long long ga = (unsigned long long)(uintptr_t)gsrc;
    v4u g0;
    g0.x = 1u;                                                 // count=1 (valid), user mode
    g0.y = ldsOff;                                             // lds_addr (bytes)
    g0.z = (unsigned)(ga & 0xFFFFFFFFu);                       // global_addr[31:0]
    g0.w = (unsigned)((ga >> 32) & 0x01FFFFFFu) | (2u << 30);  // global_addr[56:32] | type=2
    v8i g1;
    g1[0] = (int)((2u << 16)        // data_size = 4 bytes
                | (1u << 20)        // pad_enable
                | (7u << 22)        // pad_interval: every 256 DWORDs
                | (3u << 25));      // pad_amount: 4 DWORDs
    g1[1] = (int)((unsigned)D_TOT << 16);                      // tensor_dim0[15:0] = 256
    g1[2] = (int)((unsigned)(B_TOT & 0xFFFF) << 16);           // tensor_dim1[15:0] = 8192
    g1[3] = (int)(((unsigned)B_TOT >> 16) | ((unsigned)D_TOT << 16)); // tdim1 hi | tile_dim0=256
    g1[4] = (int)(tileRows & 0xFFFFu);                         // tile_dim1 | tile_dim2=0
    g1[5] = (int)D_TOT;                                        // tensor_dim0_stride = 256
    g1[6] = 0;                                                 // stride hi | tdim1_stride lo
    g1[7] = 0;
    v4i z4 = (v4i){0, 0, 0, 0};
    v8i z8 = (v8i){0, 0, 0, 0, 0, 0, 0, 0};
    __builtin_amdgcn_tensor_load_to_lds(g0, g1, z4, z4, z8, 0);
}

// ---------------------------------------------------------------------------
// Kernel 1: t2[j] = ||target_j||^2   (one wave per row)
// ---------------------------------------------------------------------------
__global__ __launch_bounds__(256) void bmc_t2_kernel(const float* __restrict__ tgt,
                                                     float* __restrict__ t2) {
    int row  = blockIdx.x * 8 + (threadIdx.x >> 5);
    int lane = threadIdx.x & 31;
    const float* p = tgt + (size_t)row * D_TOT;
    float s = 0.0f;
    for (int k = lane; k < D_TOT; k += 32) {
        float v = p[k];
        s = fmaf(v, v, s);
    }
    #pragma unroll
    for (int off = 16; off > 0; off >>= 1) s += __shfl_xor(s, off, 32);
    if (lane == 0) t2[row] = s;
}

// ---------------------------------------------------------------------------
// Kernel 2: fused f32-WMMA GEMM (pred @ target^T) + streaming logsumexp.
//   grid = (64 row blocks, 16 col splits), 256 threads = 8 waves (wave32).
//   A panel (128x256) TDM'd into LDS once; B tiles (64x256) TDM double-
//   buffered, overlapping DMA of tile it+1 with WMMA compute of tile it.
// ---------------------------------------------------------------------------
__global__ __launch_bounds__(256) void bmc_main_kernel(
    const float* __restrict__ pred, const float* __restrict__ tgt,
    const float* __restrict__ sigma_p, const float* __restrict__ t2,
    float2* __restrict__ partials,  // [B_TOT][SPLITS]
    float* __restrict__ diag)       // [B_TOT] (log2 domain)
{
    extern __shared__ float lds[];
    float* ldsA = lds;

    const int rb   = blockIdx.x;          // row block 0..63
    const int cs   = blockIdx.y;          // col split 0..15
    const int tid  = threadIdx.x;
    const int lane = tid & 31;
    const int wave = tid >> 5;            // 0..7
    const int wy   = wave >> 1;           // 0..3 : 32-row subtile
    const int wx   = wave & 1;            // 0..1 : 32-col subtile
    const int llo  = lane & 15;
    const int lhi  = lane >> 4;

    const float sigma   = sigma_p[0];
    const float var     = sigma * sigma;
    const float s_scale = LOG2E / var;    // s' = (cross - 0.5*t2) * s_scale  (log2 domain)

    // prologue: DMA A panel + first B tile (wave 0 issues; TENSORcnt tracks)
    if (wave == 0) {
        tdm_load_2d_to_lds(LDSA_OFF, pred + (size_t)rb * MBLK * D_TOT, MBLK);
        tdm_load_2d_to_lds(LDSB0_OFF, tgt + (size_t)(cs * COLS_PER_WG) * D_TOT, NBLK);
    }

    // per-lane online stats: rows m*16 + r + 8*lhi  (m in 0..1, r in 0..7)
    float rmax[2][8], rsum[2][8];
    #pragma unroll
    for (int m = 0; m < 2; ++m)
        #pragma unroll
        for (int r = 0; r < 8; ++r) { rmax[m][r] = -1.0e30f; rsum[m][r] = 0.0f; }

    const int row_base = rb * MBLK + wy * 32 + 8 * lhi;

    for (int it = 0; it < NITER; ++it) {
        const int col0 = cs * COLS_PER_WG + it * NBLK;

        __syncthreads();   // all waves done reading the buffer TDM is about to refill
        if (wave == 0) {
            if (it + 1 < NITER) {
                tdm_load_2d_to_lds(((it + 1) & 1) ? LDSB1_OFF : LDSB0_OFF,
                                   tgt + (size_t)(col0 + NBLK) * D_TOT, NBLK);
                __builtin_amdgcn_s_wait_tensorcnt(1);  // in-order: tile `it` has landed
            } else {
                __builtin_amdgcn_s_wait_tensorcnt(0);
            }
        }
        __syncthreads();   // tile `it` (and A on it==0) visible to all waves

        // double-buffer select via plain pointer arithmetic (no pointer array:
        // an addrspace(3)-derived pointer array breaks static initializers)
        const float* ldsB = lds + (MBLK + (it & 1) * NBLK) * PADROW;

        v8f acc[2][2];
        #pragma unroll
        for (int m = 0; m < 2; ++m)
            #pragma unroll
            for (int n = 0; n < 2; ++n)
                #pragma unroll
                for (int e = 0; e < 8; ++e) acc[m][n][e] = 0.0f;

        // fragment base pointers: lane holds K pair (k + 2*lhi, +1) of its row/col
        const float* aB0 = ldsA + (wy * 32 + llo) * PADROW + lhi * 2;
        const float* aB1 = aB0 + 16 * PADROW;
        const float* bB0 = ldsB + (wx * 32 + llo) * PADROW + lhi * 2;
        const float* bB1 = bB0 + 16 * PADROW;

        #pragma unroll 8
        for (int k = 0; k < D_TOT; k += 4) {
            v2f a0 = *(const v2f*)(aB0 + k);
            v2f a1 = *(const v2f*)(aB1 + k);
            v2f b0 = *(const v2f*)(bB0 + k);
            v2f b1 = *(const v2f*)(bB1 + k);
            acc[0][0] = __builtin_amdgcn_wmma_f32_16x16x4_f32(false, a0, false, b0, (short)0, acc[0][0], false, false);
            acc[0][1] = __builtin_amdgcn_wmma_f32_16x16x4_f32(false, a0, false, b1, (short)0, acc[0][1], false, false);
            acc[1][0] = __builtin_amdgcn_wmma_f32_16x16x4_f32(false, a1, false, b0, (short)0, acc[1][0], false, false);
            acc[1][1] = __builtin_amdgcn_wmma_f32_16x16x4_f32(false, a1, false, b1, (short)0, acc[1][1], false, false);
        }

        // epilogue: fold this 32x32 wave tile into online stats (log2 domain)
        const int j0 = col0 + wx * 32 + llo;   // ntile 0 column of this lane
        const int j1 = j0 + 16;                // ntile 1 column
        const float off0 = -0.5f * t2[j0] * s_scale;
        const float off1 = -0.5f * t2[j1] * s_scale;

        #pragma unroll
        for (int m = 0; m < 2; ++m) {
            const int rowg = row_base + m * 16;
            #pragma unroll
            for (int r = 0; r < 8; ++r) {
                const int row = rowg + r;
                float v0 = fmaf(acc[m][0][r], s_scale, off0);
                float v1 = fmaf(acc[m][1][r], s_scale, off1);
                if (j0 == row) diag[row] = v0;
                if (j1 == row) diag[row] = v1;
                float mx = fmaxf(v0, v1);
                float nm = fmaxf(rmax[m][r], mx);
                rsum[m][r] = rsum[m][r] * fexp2(rmax[m][r] - nm)
                           + fexp2(v0 - nm) + fexp2(v1 - nm);
                rmax[m][r] = nm;
            }
        }
    }

    // cross-lane merge: 16 lanes (per wave half) share each row
    #pragma unroll
    for (int m = 0; m < 2; ++m) {
        #pragma unroll
        for (int r = 0; r < 8; ++r) {
            float mmax = rmax[m][r];
            float msum = rsum[m][r];
            #pragma unroll
            for (int off = 1; off < 16; off <<= 1) {
                float omax = __shfl_xor(mmax, off, 32);
                float osum = __shfl_xor(msum, off, 32);
                float nm = fmaxf(mmax, omax);
                msum = msum * fexp2(mmax - nm) + osum * fexp2(omax - nm);
                mmax = nm;
            }
            if (llo == 0) {
                int row = row_base + m * 16 + r;
                partials[(size_t)row * SPLITS + cs] = make_float2(mmax, msum);
            }
        }
    }
}

// ---------------------------------------------------------------------------
// Kernel 3: merge per-split partials, subtract diag, mean, scale by 2*var.
// ---------------------------------------------------------------------------
__global__ __launch_bounds__(256) void bmc_reduce_kernel(
    const float2* __restrict__ partials, const float* __restrict__ diag,
    const float* __restrict__ sigma_p, float* __restrict__ out)
{
    __shared__ float red[256];
    float acc = 0.0f;
    for (int row = threadIdx.x; row < B_TOT; row += 256) {
        float mmax = -1.0e30f, msum = 0.0f;
        #pragma unroll
        for (int s = 0; s < SPLITS; ++s) {
            float2 p = partials[(size_t)row * SPLITS + s];
            float nm = fmaxf(mmax, p.x);
            msum = msum * fexp2(mmax - nm) + p.y * fexp2(p.x - nm);
            mmax = nm;
        }
        float lse2 = mmax + log2f(msum);
        acc += (lse2 - diag[row]);
    }
    red[threadIdx.x] = acc;
    __syncthreads();
    for (int s = 128; s > 0; s >>= 1) {
        if (threadIdx.x < s) red[threadIdx.x] += red[threadIdx.x + s];
        __syncthreads();
    }
    if (threadIdx.x == 0) {
        float sigma = sigma_p[0];
        float var = sigma * sigma;
        out[0] = LN2 * (red[0] / (float)B_TOT) * 2.0f * var;
    }
}

// ---------------------------------------------------------------------------
extern "C" void kernel_launch(void* const* d_in, const int* in_sizes, int n_in,
                              void* d_out, int out_size, void* d_ws, size_t ws_size,
                              hipStream_t stream) {
    (void)in_sizes; (void)n_in; (void)out_size; (void)ws_size;
    const float* pred  = (const float*)d_in[0];
    const float* tgt   = (const float*)d_in[1];
    const float* sigma = (const float*)d_in[2];
    float* out = (float*)d_out;

    // workspace layout (floats): t2[8192] | diag[8192] | partials[8192][16]{max,sum}
    float*  t2       = (float*)d_ws;
    float*  diag     = t2 + B_TOT;
    float2* partials = (float2*)(diag + B_TOT);

    bmc_t2_kernel<<<B_TOT / 8, 256, 0, stream>>>(tgt, t2);

    dim3 grid(B_TOT / MBLK, SPLITS);
    bmc_main_kernel<<<grid, 256, LDS_TOTAL, stream>>>(pred, tgt, sigma, t2, partials, diag);

    bmc_reduce_kernel<<<1, 256, 0, stream>>>(partials, diag, sigma, out);
}